// Voxelization_24764781429164
// MI455X (gfx1250) — compile-verified
//
#include <hip/hip_runtime.h>
#include <hip/hip_bf16.h>
#include <math.h>

typedef __attribute__((ext_vector_type(2))) float v2f;
typedef __attribute__((ext_vector_type(8))) float v8f;

#define Bcnt 16
#define Dch  64
#define Npts 32768
#define RES  32
#define R3   32768   // 32^3, happens to equal Npts

// ---------------------------------------------------------------- zero (float4)
__global__ void vox_zero_kernel(float4* p, int n4) {
    int i = blockIdx.x * blockDim.x + threadIdx.x;
    if (i < n4) p[i] = make_float4(0.f, 0.f, 0.f, 0.f);
}

// ---------------------------------------------------------------- per-batch stats
// One 256-thread block per batch. Channel sums via V_WMMA_F32_16X16X4_F32 with
// B = ones: D(M,N) += sum_k A(M,k). Element->(M,k) mapping is irrelevant because
// we reduce over everything. Column 0 of C lives in lane0 (M=0..7) and lane16
// (M=8..15), so sum-of-8-VGPRs + shfl_xor(16) yields the wave total in all lanes.
__global__ void vox_stats_kernel(const float* __restrict__ coords,
                                 float* __restrict__ stats) {
    const int b    = blockIdx.x;
    const int lane = threadIdx.x & 31;
    const int wave = threadIdx.x >> 5;           // 8 waves
    __shared__ float red[8];
    __shared__ float means[3];

    // ---- pass 1: channel sums -> mean
    for (int c = 0; c < 3; ++c) {
        const float* p = coords + ((size_t)b * 3 + c) * Npts;
        const int start = wave * (Npts / 8);     // 4096 elems per wave
        v8f acc = {};
        v2f bones; bones.x = 1.0f; bones.y = 1.0f;
        for (int chunk = 0; chunk < Npts / 8; chunk += 64) {
            v2f a;
            a.x = p[start + chunk + lane];
            a.y = p[start + chunk + 32 + lane];
            acc = __builtin_amdgcn_wmma_f32_16x16x4_f32(
                false, a, false, bones, (short)0, acc, false, false);
        }
        float s = acc[0] + acc[1] + acc[2] + acc[3] +
                  acc[4] + acc[5] + acc[6] + acc[7];
        s += __shfl_xor(s, 16, 32);              // combine M=0..7 with M=8..15
        if (lane == 0) red[wave] = s;
        __syncthreads();
        if (threadIdx.x == 0) {
            float t = 0.f;
            for (int w = 0; w < 8; ++w) t += red[w];
            means[c] = t / (float)Npts;
        }
        __syncthreads();
    }
    const float mx = means[0], my = means[1], mz = means[2];

    // ---- pass 2: max squared norm of centered coords
    const float* px = coords + ((size_t)b * 3 + 0) * Npts;
    const float* py = coords + ((size_t)b * 3 + 1) * Npts;
    const float* pz = coords + ((size_t)b * 3 + 2) * Npts;
    float m = 0.f;
    for (int n = threadIdx.x; n < Npts; n += 256) {
        float dx = px[n] - mx, dy = py[n] - my, dz = pz[n] - mz;
        m = fmaxf(m, dx * dx + dy * dy + dz * dz);
    }
    for (int off = 16; off > 0; off >>= 1)
        m = fmaxf(m, __shfl_xor(m, off, 32));
    if (lane == 0) red[wave] = m;
    __syncthreads();
    if (threadIdx.x == 0) {
        float t = red[0];
        for (int w = 1; w < 8; ++w) t = fmaxf(t, red[w]);
        float inv = 1.0f / (2.0f * sqrtf(t));    // EPS == 0
        stats[b * 4 + 0] = mx;
        stats[b * 4 + 1] = my;
        stats[b * 4 + 2] = mz;
        stats[b * 4 + 3] = inv;
    }
}

// ---------------------------------------------------------------- scatter
__global__ void vox_scatter_kernel(const float* __restrict__ feat,
                                   const float* __restrict__ coords,
                                   const float* __restrict__ stats,
                                   float* __restrict__ out_vox,
                                   float* __restrict__ out_nc,
                                   float* __restrict__ counts) {
    const int gid = blockIdx.x * blockDim.x + threadIdx.x;   // B*N threads
    const int b = gid >> 15;
    const int n = gid & (Npts - 1);

    const float mx  = stats[b * 4 + 0];
    const float my  = stats[b * 4 + 1];
    const float mz  = stats[b * 4 + 2];
    const float inv = stats[b * 4 + 3];

    const float* cb = coords + (size_t)b * 3 * Npts;
    float x = (cb[n]             - mx) * inv + 0.5f;
    float y = (cb[Npts + n]      - my) * inv + 0.5f;
    float z = (cb[2 * Npts + n]  - mz) * inv + 0.5f;
    float ncx = fminf(fmaxf(x * (float)RES, 0.f), (float)(RES - 1));
    float ncy = fminf(fmaxf(y * (float)RES, 0.f), (float)(RES - 1));
    float ncz = fminf(fmaxf(z * (float)RES, 0.f), (float)(RES - 1));

    out_nc[((size_t)b * 3 + 0) * Npts + n] = ncx;
    out_nc[((size_t)b * 3 + 1) * Npts + n] = ncy;
    out_nc[((size_t)b * 3 + 2) * Npts + n] = ncz;

    const int vx = (int)rintf(ncx);              // RNE matches jnp.round
    const int vy = (int)rintf(ncy);
    const int vz = (int)rintf(ncz);
    const int idx = vx * (RES * RES) + vy * RES + vz;

    atomicAdd(&counts[(size_t)b * R3 + idx], 1.0f);

    const float* fb = feat    + (size_t)b * Dch * Npts + n;    // coalesced per d
    float*       ob = out_vox + (size_t)b * Dch * R3   + idx;  // L2-resident RMW
    #pragma unroll 8
    for (int d = 0; d < Dch; ++d) {
        if (d + 8 < Dch) __builtin_prefetch(fb + (size_t)(d + 8) * Npts, 0, 0);
        atomicAdd(ob + (size_t)d * R3, fb[(size_t)d * Npts]);
    }
}

// ---------------------------------------------------------------- divide (float4)
__global__ void vox_divide_kernel(float* __restrict__ out_vox,
                                  const float* __restrict__ counts, int n4) {
    int i4 = blockIdx.x * blockDim.x + threadIdx.x;
    if (i4 >= n4) return;
    size_t i = (size_t)i4 * 4;
    const int v = (int)(i & (R3 - 1));           // 4 consecutive v, same (b,d)
    const int b = (int)((i >> 15) >> 6);
    float4 s = ((float4*)out_vox)[i4];
    const float4 c = *(const float4*)(counts + (size_t)b * R3 + v);
    s.x /= fmaxf(c.x, 1.f);
    s.y /= fmaxf(c.y, 1.f);
    s.z /= fmaxf(c.z, 1.f);
    s.w /= fmaxf(c.w, 1.f);
    ((float4*)out_vox)[i4] = s;
}

extern "C" void kernel_launch(void* const* d_in, const int* in_sizes, int n_in,
                              void* d_out, int out_size, void* d_ws, size_t ws_size,
                              hipStream_t stream) {
    const float* feat   = (const float*)d_in[0];   // (B, D, N)
    const float* coords = (const float*)d_in[1];   // (B, 3, N)
    float* out     = (float*)d_out;
    float* out_vox = out;                                   // B*D*R3 floats
    float* out_nc  = out + (size_t)Bcnt * Dch * R3;         // B*3*N floats
    float* counts  = (float*)d_ws;                          // B*R3 floats
    float* stats   = counts + (size_t)Bcnt * R3;            // B*4 floats

    // zero voxel accumulator (in d_out) and counts+stats (in d_ws)
    {
        int n4 = (Bcnt * Dch * R3) / 4;                     // 8,388,608
        vox_zero_kernel<<<(n4 + 255) / 256, 256, 0, stream>>>((float4*)out_vox, n4);
        int c4 = (Bcnt * R3 + Bcnt * 4) / 4;                // counts + stats
        vox_zero_kernel<<<(c4 + 255) / 256, 256, 0, stream>>>((float4*)counts, c4);
    }

    vox_stats_kernel<<<Bcnt, 256, 0, stream>>>(coords, stats);

    vox_scatter_kernel<<<(Bcnt * Npts) / 256, 256, 0, stream>>>(
        feat, coords, stats, out_vox, out_nc, counts);

    {
        int n4 = (Bcnt * Dch * R3) / 4;
        vox_divide_kernel<<<(n4 + 255) / 256, 256, 0, stream>>>(out_vox, counts, n4);
    }
}